// Select_62466004353051
// MI455X (gfx1250) — compile-verified
//
#include <hip/hip_runtime.h>

typedef float v2f __attribute__((ext_vector_type(2)));
typedef float v8f __attribute__((ext_vector_type(8)));

#define Bsz 128
#define Rr  64
#define Ww  64
#define Dd  128

#define SAB 68   // 64 + 4 pad floats: 16B-aligned rows (b128 copies) + bank stagger
#define SS  65   // S tile stride: conflict-free row AND column access

// async copy operand types (pointee width per compiler diagnostic probing)
typedef int v4i __attribute__((vector_size(4 * sizeof(int))));
typedef __attribute__((address_space(1))) v4i gbl_v4i;
typedef __attribute__((address_space(3))) v4i lds_v4i;

__device__ __forceinline__ void async_copy_f4(const float* g, float* l) {
#if __has_builtin(__builtin_amdgcn_global_load_async_to_lds_b128)
  __builtin_amdgcn_global_load_async_to_lds_b128((gbl_v4i*)g, (lds_v4i*)l, 0, 0);
#else
  *(float4*)l = *(const float4*)g;
#endif
}

__device__ __forceinline__ void wait_async_zero() {
#if __has_builtin(__builtin_amdgcn_s_wait_asynccnt)
  __builtin_amdgcn_s_wait_asynccnt(0);
#else
  asm volatile("s_wait_asynccnt 0x0" ::: "memory");
#endif
}

__global__ __launch_bounds__(128)
void select_sparsemax_kernel(const float* __restrict__ imgs,
                             const float* __restrict__ caps,
                             const int* __restrict__ img_lens,
                             const int* __restrict__ cap_lens,
                             float* __restrict__ out) {
  __shared__ float sA[2][Rr * SAB];
  __shared__ float sB[2][Ww * SAB];
  __shared__ float sS[Rr * SS];
  __shared__ float sAcc[2];

  const int i    = blockIdx.x;
  const int t    = blockIdx.y;
  const int tid  = threadIdx.x;
  const int lane = tid & 31;
  const int wave = tid >> 5;

  const int lenI = img_lens[i];
  const int lenT = cap_lens[t];

  if (tid < 2) sAcc[tid] = 0.0f;

  v8f acc[4];
  v8f vzero = {0.f, 0.f, 0.f, 0.f, 0.f, 0.f, 0.f, 0.f};
#pragma unroll
  for (int cb = 0; cb < 4; ++cb) acc[cb] = vzero;

  const float* gA = imgs + (size_t)i * Rr * Dd;
  const float* gB = caps + (size_t)t * Ww * Dd;

  const int mrow = lane & 15;   // M (A) / N (B,C) index within 16-wide tile
  const int half = lane >> 4;   // K-pair select (A/B) / M+8 rows (C)

  // ---- stage one K-half (64 rows x 64 floats each matrix) into buffer `buf`
  auto stage = [&](int buf, int kh) {
    for (int e = tid; e < Rr * 16; e += 128) {  // 16 float4 per row-half
      int row = e >> 4;
      int c4  = e & 15;
      async_copy_f4(gA + (size_t)row * Dd + kh * 64 + 4 * c4,
                    &sA[buf][row * SAB + 4 * c4]);
      async_copy_f4(gB + (size_t)row * Dd + kh * 64 + 4 * c4,
                    &sB[buf][row * SAB + 4 * c4]);
    }
  };

  // ---- 16x16 f32 WMMA over one staged K-half ----
  auto gemm_half = [&](int buf) {
    const float* aRow = &sA[buf][(wave * 16 + mrow) * SAB + 2 * half];
#pragma unroll 4
    for (int k0 = 0; k0 < 64; k0 += 4) {
      v2f a;
      a.x = aRow[k0];
      a.y = aRow[k0 + 1];
#pragma unroll
      for (int cb = 0; cb < 4; ++cb) {
        const float* bRow = &sB[buf][(cb * 16 + mrow) * SAB + 2 * half];
        v2f b;
        b.x = bRow[k0];
        b.y = bRow[k0 + 1];
        acc[cb] = __builtin_amdgcn_wmma_f32_16x16x4_f32(
            false, a, false, b, (short)0, acc[cb], false, false);
      }
    }
  };

  // ---- double-buffered pipeline: copy half1 flies during half0 compute ----
  stage(0, 0);
  wait_async_zero();
  __syncthreads();

  stage(1, 1);          // async: overlaps with gemm_half(0)
  gemm_half(0);

  wait_async_zero();
  __syncthreads();
  gemm_half(1);

  // ---- spill S tiles to LDS (C layout: vgpr v -> M=v+8*half, N=lane&15) ----
#pragma unroll
  for (int cb = 0; cb < 4; ++cb) {
    int col = cb * 16 + mrow;
#pragma unroll
    for (int v = 0; v < 8; ++v) {
      int m = v + 8 * half;
      sS[(wave * 16 + m) * SS + col] = acc[cb][v];
    }
  }
  __syncthreads();

  // ---- sparsemax lines: 0..63 = rows (r2w), 64..127 = cols (w2r) ----
  for (int line = wave; line < 128; line += 4) {
    const bool isRow = line < 64;
    const int  li    = isRow ? line : line - 64;
    if (isRow ? (li >= lenI) : (li >= lenT)) continue;  // zero marginal weight

    const int idx0 = 2 * lane, idx1 = 2 * lane + 1;
    float z0, z1;
    if (isRow) {
      z0 = (idx0 < lenT) ? sS[li * SS + idx0] : -1.0f;
      z1 = (idx1 < lenT) ? sS[li * SS + idx1] : -1.0f;
    } else {
      z0 = (idx0 < lenI) ? sS[idx0 * SS + li] : -1.0f;
      z1 = (idx1 < lenI) ? sS[idx1 * SS + li] : -1.0f;
    }

    // bitonic sort, descending, 64 virtual elements (2 slots/lane, wave32)
#pragma unroll
    for (int k = 2; k <= 64; k <<= 1) {
#pragma unroll
      for (int j = k >> 1; j >= 1; j >>= 1) {
        const bool desc = ((lane & (k >> 1)) == 0);
        if (j == 1) {                         // intra-lane (slot) exchange
          float hi = fmaxf(z0, z1), lo = fminf(z0, z1);
          z0 = desc ? hi : lo;
          z1 = desc ? lo : hi;
        } else {                              // cross-lane exchange
          const int lj = j >> 1;
          float o0 = __shfl_xor(z0, lj, 32);
          float o1 = __shfl_xor(z1, lj, 32);
          const bool keepMax = (desc == ((lane & lj) == 0));
          z0 = keepMax ? fmaxf(z0, o0) : fminf(z0, o0);
          z1 = keepMax ? fmaxf(z1, o1) : fminf(z1, o1);
        }
      }
    }

    // inclusive scans of z and z^2 over sorted order
    float ps = z0 + z1;
    float qs = z0 * z0 + z1 * z1;
    float pi = ps, qi = qs;
#pragma unroll
    for (int d = 1; d < 32; d <<= 1) {
      float pa = __shfl_up(pi, d, 32);
      float qa = __shfl_up(qi, d, 32);
      if (lane >= d) { pi += pa; qi += qa; }
    }
    float c1 = pi;                 // cumsum(z)   at idx 2l+1
    float c0 = pi - ps + z0;       // cumsum(z)   at idx 2l
    float q1 = qi;                 // cumsum(z^2) at idx 2l+1
    float q0 = qi - qs + z0 * z0;  // cumsum(z^2) at idx 2l

    // support size k: count of (z_sorted * m > cumsum_m - 1)
    bool cond0 = z0 * (float)(idx0 + 1) > c0 - 1.0f;
    bool cond1 = z1 * (float)(idx1 + 1) > c1 - 1.0f;
    int k = __popcll(__ballot(cond0)) + __popcll(__ballot(cond1));

    int kl = (k - 1) >> 1;
    int ks = (k - 1) & 1;
    float ck0 = __shfl(c0, kl, 32), ck1 = __shfl(c1, kl, 32);
    float qk0 = __shfl(q0, kl, 32), qk1 = __shfl(q1, kl, 32);
    float ck = ks ? ck1 : ck0;
    float qk = ks ? qk1 : qk0;
    float tau = (ck - 1.0f) / (float)k;
    // (sparsemax(z) * z).sum() == sum_{m<=k} z_m^2 - tau * sum_{m<=k} z_m
    float contrib = qk - tau * ck;

    if (lane == 0) atomicAdd(&sAcc[isRow ? 0 : 1], contrib);
  }

  __syncthreads();
  if (tid == 0) {
    out[(size_t)i * Bsz + t] =
        0.5f * (sAcc[0] / (float)lenI + sAcc[1] / (float)lenT);
  }
}

extern "C" void kernel_launch(void* const* d_in, const int* in_sizes, int n_in,
                              void* d_out, int out_size, void* d_ws, size_t ws_size,
                              hipStream_t stream) {
  // setup_inputs order: img_cls, imgs, cap_cls, caps, img_lens, cap_lens
  const float* imgs     = (const float*)d_in[1];
  const float* caps     = (const float*)d_in[3];
  const int*   img_lens = (const int*)d_in[4];
  const int*   cap_lens = (const int*)d_in[5];
  float*       out      = (float*)d_out;

  dim3 grid(Bsz, Bsz);
  hipLaunchKernelGGL(select_sparsemax_kernel, grid, dim3(128), 0, stream,
                     imgs, caps, img_lens, cap_lens, out);
}